// Schrodinger1D_2559800508818
// MI455X (gfx1250) — compile-verified
//
#include <hip/hip_runtime.h>
#include <hip/hip_bf16.h>

#define NMESH   4096
#define THREADS 1024
#define NWAVES  (THREADS / 32)
#define NITER   14
#define SIGMA   0.5f

typedef __attribute__((ext_vector_type(2))) float v2f;
typedef __attribute__((ext_vector_type(8))) float v8f;

// Block-wide sum of 4096 values (4 per thread) using V_WMMA_F32_16X16X4_F32.
// A = ones(16x4), B = data(4x16): C[m][n] += colsum(n). Total sum is layout-
// agnostic. c[0] holds C[0][n] (lanes 0-15) and C[8][n] (lanes 16-31), both
// equal to column sums, so the 32-lane sum of c[0] = 2 * (wave's 128-value sum).
__device__ __forceinline__ float block_sum_wmma(float v0, float v1, float v2, float v3,
                                                float* s_red) {
    v2f ones; ones.x = 1.0f; ones.y = 1.0f;
    v2f b0;   b0.x = v0;     b0.y = v1;
    v2f b1;   b1.x = v2;     b1.y = v3;
    v8f c = {};
    c = __builtin_amdgcn_wmma_f32_16x16x4_f32(false, ones, false, b0, (short)0, c, false, false);
    c = __builtin_amdgcn_wmma_f32_16x16x4_f32(false, ones, false, b1, (short)0, c, false, false);
    float partial = c[0];
    #pragma unroll
    for (int off = 16; off >= 1; off >>= 1)
        partial += __shfl_xor(partial, off, 32);
    partial *= 0.5f;                       // rows 0 and 8 both counted

    __syncthreads();                       // protect s_red reuse across calls
    if ((threadIdx.x & 31) == 0) s_red[threadIdx.x >> 5] = partial;
    __syncthreads();
    float tot = 0.0f;
    #pragma unroll
    for (int w = 0; w < NWAVES; ++w) tot += s_red[w];   // identical order -> deterministic
    return tot;
}

__global__ __launch_bounds__(THREADS)
void Schrodinger1D_kernel(const float* __restrict__ target,
                          const float* __restrict__ potential,
                          const float* __restrict__ K,
                          float* __restrict__ out) {
    // LDS-resident state: 8 x 16KB = 131KB (< 320KB / WGP on CDNA5)
    __shared__ float s_a[NMESH], s_b[NMESH], s_c[NMESH], s_f[NMESH];
    __shared__ float s_d[NMESH], s_al[NMESH], s_cu[NMESH], s_p[NMESH];
    __shared__ float s_red[NWAVES];

    const int tid = threadIdx.x;

    // CDNA5 prefetch (global_prefetch_b8): warm target for the final pass.
    __builtin_prefetch(&target[tid], 0, 1);
    __builtin_prefetch(&target[tid + 2048], 0, 1);

    // H is tridiagonal: read only the band of K (everything else is zero).
    for (int i = tid; i < NMESH; i += THREADS) {
        size_t row = (size_t)i * NMESH;
        s_d[i]  = potential[i] + K[row + i];
        s_al[i] = (i > 0)         ? K[row + i - 1] : 0.0f;
        s_cu[i] = (i < NMESH - 1) ? K[row + i + 1] : 0.0f;
        s_p[i]  = 1.0f;                     // initial guess (positive ground state)
    }
    __syncthreads();

    // Shifted inverse iteration: solve (H - sigma I) x = psi, normalize.
    for (int it = 0; it < NITER; ++it) {
        // rebuild tridiagonal system
        for (int i = tid; i < NMESH; i += THREADS) {
            s_a[i] = s_al[i];
            s_c[i] = s_cu[i];
            s_b[i] = s_d[i] - SIGMA;
            s_f[i] = s_p[i];
        }
        __syncthreads();

        // cyclic reduction: forward elimination (in place)
        for (int s = 1; s < NMESH; s <<= 1) {
            const int stride2 = s << 1;
            const int nrows   = NMESH / stride2;
            for (int r = tid; r < nrows; r += THREADS) {
                const int i  = stride2 * (r + 1) - 1;
                const int im = i - s;
                const int ip = i + s;
                const float alpha = s_a[i] / s_b[im];
                float an = -alpha * s_a[im];
                float bn = s_b[i] - alpha * s_c[im];
                float fn = s_f[i] - alpha * s_f[im];
                float cn = 0.0f;
                if (ip < NMESH) {
                    const float beta = s_c[i] / s_b[ip];
                    bn -= beta * s_a[ip];
                    fn -= beta * s_f[ip];
                    cn  = -beta * s_c[ip];
                }
                s_a[i] = an; s_b[i] = bn; s_c[i] = cn; s_f[i] = fn;
            }
            __syncthreads();
        }

        // back substitution, in place in s_f (each row solved exactly once;
        // neighbors at distance s were solved at strictly larger levels)
        for (int s = NMESH; s >= 1; s >>= 1) {
            const int stride2 = s << 1;
            const int nrows   = (NMESH - (s - 1) + stride2 - 1) / stride2;
            for (int r = tid; r < nrows; r += THREADS) {
                const int i  = (s - 1) + stride2 * r;
                const float xm = (i - s >= 0)    ? s_f[i - s] : 0.0f;
                const float xp = (i + s < NMESH) ? s_f[i + s] : 0.0f;
                s_f[i] = (s_f[i] - s_a[i] * xm - s_c[i] * xp) / s_b[i];
            }
            __syncthreads();
        }

        // normalize: ||x||^2 via WMMA reduction
        const float x0 = s_f[tid];
        const float x1 = s_f[tid + THREADS];
        const float x2 = s_f[tid + 2 * THREADS];
        const float x3 = s_f[tid + 3 * THREADS];
        const float nrm2 = block_sum_wmma(x0 * x0, x1 * x1, x2 * x2, x3 * x3, s_red);
        const float inv  = rsqrtf(nrm2);
        for (int i = tid; i < NMESH; i += THREADS) s_p[i] = s_f[i] * inv;
        __syncthreads();
    }

    // residual: sum |psi^2 - target|  (sign of eigenvector irrelevant)
    float r[4];
    #pragma unroll
    for (int j = 0; j < 4; ++j) {
        const int i = tid + j * THREADS;
        const float p = s_p[i];
        r[j] = fabsf(p * p - target[i]);
    }
    const float res = block_sum_wmma(r[0], r[1], r[2], r[3], s_red);
    if (tid == 0) out[0] = res;
}

extern "C" void kernel_launch(void* const* d_in, const int* in_sizes, int n_in,
                              void* d_out, int out_size, void* d_ws, size_t ws_size,
                              hipStream_t stream) {
    const float* target    = (const float*)d_in[0];
    const float* potential = (const float*)d_in[1];
    const float* K         = (const float*)d_in[2];
    float*       out       = (float*)d_out;
    (void)in_sizes; (void)n_in; (void)out_size; (void)d_ws; (void)ws_size;

    hipLaunchKernelGGL(Schrodinger1D_kernel, dim3(1), dim3(THREADS), 0, stream,
                       target, potential, K, out);
}